// PerspectiveWarp_20564303414044
// MI455X (gfx1250) — compile-verified
//
#include <hip/hip_runtime.h>

#define HH 384
#define WW 1280
#define BB 8
#define KK 16
#define PX 4              // pixels per thread
#define TPB (WW / PX)     // 320 threads = 10 wave32 = exactly one image row

// Two adjacent floats, 4-byte aligned: backend fuses to global_load_b64
// on gfx1250's unaligned-VMEM path (confirmed by round-1 histogram).
struct __attribute__((packed, aligned(4))) f2a4 { float lo, hi; };

typedef float v4f __attribute__((ext_vector_type(4)));

__global__ __launch_bounds__(TPB) void PerspectiveWarp_kernel(
    const float* __restrict__ cam1, const float* __restrict__ cam2,
    const float* __restrict__ im2,  const float* __restrict__ d1,
    const float* __restrict__ rotm, const float* __restrict__ t,
    float* __restrict__ out_warp,   float* __restrict__ out_mask)
{
    // One block == one image row of one batch: b, h are pure SGPR values.
    const int b  = blockIdx.x / HH;
    const int h  = blockIdx.x - b * HH;
    const int w0 = threadIdx.x * PX;

    // Per-batch parameters: uniform offsets -> one s_load burst.
    const float fx1 = cam1[b*4+0], fy1 = cam1[b*4+1];
    const float cx1 = cam1[b*4+2], cy1 = cam1[b*4+3];
    const float fx2 = cam2[b*4+0], fy2 = cam2[b*4+1];
    const float cx2 = cam2[b*4+2], cy2 = cam2[b*4+3];
    const float r00 = rotm[b*9+0], r01 = rotm[b*9+1], r02 = rotm[b*9+2];
    const float r10 = rotm[b*9+3], r11 = rotm[b*9+4], r12 = rotm[b*9+5];
    const float r20 = rotm[b*9+6], r21 = rotm[b*9+7], r22 = rotm[b*9+8];
    const float t0  = t[b*3+0],    t1  = t[b*3+1],    t2  = t[b*3+2];

    // One reciprocal each instead of 4 (resp. 1) IEEE divides.
    const float invfx1 = 1.0f / fx1;
    const float invfy1 = 1.0f / fy1;
    const float yray   = ((float)h - cy1) * invfy1;   // row-uniform ray y

    // Depth for 4 consecutive pixels: single NT b128 load (16B aligned).
    const v4f dv = __builtin_nontemporal_load(
        (const v4f*)(d1 + (size_t)(b * HH + h) * WW + w0));
    const float dd[PX] = {dv.x, dv.y, dv.z, dv.w};

    float wx[PX], wy[PX], mk[PX];
    int   off[PX];

    #pragma unroll
    for (int i = 0; i < PX; ++i) {
        const float xr = ((float)(w0 + i) - cx1) * invfx1;
        const float X  = fmaf(r00, xr, fmaf(r01, yray, fmaf(dd[i], t0, r02)));
        const float Y  = fmaf(r10, xr, fmaf(r11, yray, fmaf(dd[i], t1, r12)));
        const float Z  = fmaf(r20, xr, fmaf(r21, yray, fmaf(dd[i], t2, r22)));

        const float invZ = 1.0f / Z;                 // one divide chain/pixel
        const float u2 = fmaf(fx2, X * invZ, cx2);
        const float v2 = fmaf(fy2, Y * invZ, cy2);

        mk[i] = ((Z > 1e-5f) && (u2 > 0.0f) && (u2 < (float)(WW - 2)) &&
                 (v2 > 0.0f) && (v2 < (float)(HH - 2))) ? 1.0f : 0.0f;

        // Border-replicate bilinear coords; NaNs collapse deterministically
        // to 0 via fmaxf/fminf semantics (mask is already 0 there).
        const float u = fminf(fmaxf(u2, 0.0f), (float)(WW - 1));
        const float v = fminf(fmaxf(v2, 0.0f), (float)(HH - 1));
        int x0 = (int)floorf(u);
        int y0 = (int)floorf(v);
        x0 = (x0 > WW - 2) ? (WW - 2) : (x0 < 0 ? 0 : x0);
        y0 = (y0 > HH - 2) ? (HH - 2) : (y0 < 0 ? 0 : y0);
        wx[i]  = u - (float)x0;      // reaches 1.0 at border == replicate
        wy[i]  = v - (float)y0;
        off[i] = y0 * WW + x0;       // per-pixel gather offset (elements)
    }

    const size_t plane = (size_t)HH * WW;
    const float* __restrict__ p = im2      + (size_t)b * KK * plane;
    float*       __restrict__ o = out_warp + (size_t)b * KK * plane
                                           + (size_t)h * WW + w0;

    #pragma unroll
    for (int k = 0; k < KK; ++k) {
        float vv[PX];
        #pragma unroll
        for (int i = 0; i < PX; ++i) {
            const f2a4 top = *(const f2a4*)(p + off[i]);        // b64 gather
            const f2a4 bot = *(const f2a4*)(p + off[i] + WW);   // b64 gather
            const float tv = fmaf(wx[i], top.hi - top.lo, top.lo);
            const float bv = fmaf(wx[i], bot.hi - bot.lo, bot.lo);
            vv[i] = fmaf(wy[i], bv - tv, tv);
        }
        const v4f val = {vv[0], vv[1], vv[2], vv[3]};
        __builtin_nontemporal_store(val, (v4f*)o);   // one NT b128 store/chan
        p += plane;
        o += plane;
    }

    const v4f mv = {mk[0], mk[1], mk[2], mk[3]};
    __builtin_nontemporal_store(mv,
        (v4f*)(out_mask + (size_t)(b * HH + h) * WW + w0));     // NT b128
}

extern "C" void kernel_launch(void* const* d_in, const int* in_sizes, int n_in,
                              void* d_out, int out_size, void* d_ws, size_t ws_size,
                              hipStream_t stream) {
    const float* cam1 = (const float*)d_in[0];
    const float* cam2 = (const float*)d_in[1];
    const float* im2  = (const float*)d_in[2];
    const float* d1   = (const float*)d_in[3];
    const float* rotm = (const float*)d_in[4];
    const float* t    = (const float*)d_in[5];

    float* out_warp = (float*)d_out;
    float* out_mask = out_warp + (size_t)BB * KK * HH * WW;  // (warped, mask)

    const int nblocks = BB * HH;   // 3072 blocks x 10 wave32
    PerspectiveWarp_kernel<<<nblocks, TPB, 0, stream>>>(
        cam1, cam2, im2, d1, rotm, t, out_warp, out_mask);
}